// WECT_84559316124419
// MI455X (gfx1250) — compile-verified
//
#include <hip/hip_runtime.h>
#include <stdint.h>

// ---------------------------------------------------------------------------
// WECT on MI455X (gfx1250, wave32)
//
// Pipeline:
//  1) init          : zero max-norm slot + global histogram (in ws)
//  2) max_norm      : max ||v||^2 via wave32 shuffle reduce, sqrt once,
//                     atomicMax on float bits (non-negative => uint order)
//  3) vidx_vertex   : V_WMMA_F32_16X16X4_F32 computes 16-vertex x 64-dir
//                     projection tiles (K padded 3->4); quantize to u8 height
//                     index table (12.8MB, L2-resident) and accumulate the
//                     vertex term into a per-workgroup 64KB LDS histogram.
//                     Full tiles take a branch-free (wave-uniform) fast path.
//  4) simplex<2>    : edges: gather 2 u8 index rows (uint4 chunks), byte max,
//                     +w into LDS histogram.
//  5) simplex<3>    : triangles: 3 rows, byte max, -w.
//  6) cumsum        : per-direction inclusive prefix sum -> d_out (64x256 f32)
// ---------------------------------------------------------------------------

typedef __attribute__((ext_vector_type(2))) float v2f;
typedef __attribute__((ext_vector_type(8))) float v8f;

#define HH 256            // heights (fixed for this problem instance)
#define DD 64             // directions
#define NBINS (HH * DD)   // 16384 bins, 64KB as f32

// ws layout (bytes):
//  [0,4)                      : uint bits of max vertex norm
//  [HIST_OFF, HIST_OFF+64KB)  : global f32 histogram
//  [VIDX_OFF, +K0*DD)         : u8 per-(vertex,dir) height index table
#define HIST_OFF 128
#define VIDX_OFF (HIST_OFF + NBINS * 4)

// ---------------------------------------------------------------------------
__global__ void wect_init_kernel(float* __restrict__ ghist,
                                 unsigned* __restrict__ maxslot) {
  int i = blockIdx.x * blockDim.x + threadIdx.x;
  if (i == 0) *maxslot = 0u;
  if (i < NBINS) ghist[i] = 0.0f;
}

// ---------------------------------------------------------------------------
__global__ void wect_max_norm_kernel(const float* __restrict__ vc, int k0,
                                     unsigned* __restrict__ maxslot) {
  const int stride = gridDim.x * blockDim.x;
  float m2 = 0.0f;  // reduce on norm^2; sqrt is monotonic
  for (int i = blockIdx.x * blockDim.x + threadIdx.x; i < k0; i += stride) {
    float x = vc[3 * i + 0], y = vc[3 * i + 1], z = vc[3 * i + 2];
    m2 = fmaxf(m2, x * x + y * y + z * z);
  }
  for (int off = 16; off > 0; off >>= 1)
    m2 = fmaxf(m2, __shfl_down(m2, off, 32));
  if ((threadIdx.x & 31) == 0) atomicMax(maxslot, __float_as_uint(sqrtf(m2)));
}

// ---------------------------------------------------------------------------
// One wave computes a 16-vertex x 64-direction projection tile with 4 WMMAs.
//
// A (16x4 f32, 2 VGPRs): lanes 0-15 hold (x,y) of vertex vb+lane (K=0,1);
//                        lanes 16-31 hold (z,0) of the same vertex (K=2,3).
// B (4x16 f32, 2 VGPRs): lanes 0-15 hold (dx,dy) of direction n=lane;
//                        lanes 16-31 hold (dz,0).
// D (16x16 f32, 8 VGPRs): VGPR r -> (M=r, N=lane) lanes 0-15,
//                                   (M=r+8, N=lane-16) lanes 16-31.
__global__ void __launch_bounds__(256)
wect_vidx_vertex_kernel(const float* __restrict__ vc,
                        const float* __restrict__ vw,
                        const float* __restrict__ dirs,
                        const unsigned* __restrict__ maxslot,
                        const int* __restrict__ nh,
                        unsigned char* __restrict__ vidx,
                        float* __restrict__ ghist, int k0) {
  __shared__ float lhist[NBINS];
  for (int i = threadIdx.x; i < NBINS; i += blockDim.x) lhist[i] = 0.0f;
  __syncthreads();

  const float maxh  = __uint_as_float(*maxslot);
  const float hf    = (float)(*nh);
  const float scale = (hf - 1.0f) / (2.0f * maxh);

  const int lane = threadIdx.x & 31;
  const int half = lane >> 4;   // 0: K=0,1 half ; 1: K=2,3 half
  const int l16  = lane & 15;

  // Load B (dirs^T) once; reused across all vertex tiles.
  v2f B[4];
#pragma unroll
  for (int g = 0; g < 4; ++g) {
    const int dcol = g * 16 + l16;
    float b0, b1;
    if (half == 0) { b0 = dirs[3 * dcol + 0]; b1 = dirs[3 * dcol + 1]; }
    else           { b0 = dirs[3 * dcol + 2]; b1 = 0.0f; }
    B[g][0] = b0; B[g][1] = b1;
  }

  const int wavesPerBlock = blockDim.x >> 5;
  const int wave   = blockIdx.x * wavesPerBlock + (threadIdx.x >> 5);
  const int nwaves = gridDim.x * wavesPerBlock;
  const int ntiles = (k0 + 15) >> 4;

  for (int t = wave; t < ntiles; t += nwaves) {
    const int vb   = t << 4;
    const bool full = (vb + 16) <= k0;  // wave-uniform

    // ---- A operand (clamped addresses; zeroed via select, no branches) ----
    const int vrow  = vb + l16;
    const int vrowc = vrow < k0 ? vrow : (k0 - 1);
    float a0, a1;
    if (half == 0) { a0 = vc[3 * vrowc + 0]; a1 = vc[3 * vrowc + 1]; }
    else           { a0 = vc[3 * vrowc + 2]; a1 = 0.0f; }
    if (vrow >= k0) { a0 = 0.0f; a1 = 0.0f; }
    v2f A; A[0] = a0; A[1] = a1;

    // ---- per-lane vertex weights, hoisted: row r+8*half, reused 4x ----
    float wv[8];
#pragma unroll
    for (int r = 0; r < 8; ++r) {
      const int vtx = vb + r + 8 * half;
      wv[r] = vw[vtx < k0 ? vtx : (k0 - 1)];
    }

    // ---- 4 back-to-back WMMAs (shared A) ----
    v8f acc[4];
#pragma unroll
    for (int g = 0; g < 4; ++g) {
      v8f c = {};
      acc[g] = __builtin_amdgcn_wmma_f32_16x16x4_f32(
          /*neg_a=*/false, A, /*neg_b=*/false, B[g],
          /*c_mod=*/(short)0, c, /*reuse_a=*/false, /*reuse_b=*/false);
    }

    // ---- quantize + scatter ----
    const size_t rowbase = (size_t)(vb + 8 * half) * DD;
    if (full) {
      // branch-free fast path (K0 % 16 == 0 => always taken)
#pragma unroll
      for (int g = 0; g < 4; ++g) {
        const int dir = g * 16 + l16;
#pragma unroll
        for (int r = 0; r < 8; ++r) {
          const float idxf = ceilf((maxh + acc[g][r]) * scale);
          int idx = (int)idxf;
          idx = idx < 0 ? 0 : (idx > HH - 1 ? HH - 1 : idx);
          vidx[rowbase + (size_t)r * DD + dir] = (unsigned char)idx;
          atomicAdd(&lhist[dir * HH + idx], wv[r]);  // ds_add_f32
        }
      }
    } else {
#pragma unroll
      for (int g = 0; g < 4; ++g) {
        const int dir = g * 16 + l16;
#pragma unroll
        for (int r = 0; r < 8; ++r) {
          const int vtx = vb + r + 8 * half;
          if (vtx < k0) {
            const float idxf = ceilf((maxh + acc[g][r]) * scale);
            int idx = (int)idxf;
            idx = idx < 0 ? 0 : (idx > HH - 1 ? HH - 1 : idx);
            vidx[(size_t)vtx * DD + dir] = (unsigned char)idx;
            atomicAdd(&lhist[dir * HH + idx], wv[r]);
          }
        }
      }
    }
  }

  __syncthreads();
  for (int i = threadIdx.x; i < NBINS; i += blockDim.x)
    atomicAdd(&ghist[i], lhist[i]);
}

// ---------------------------------------------------------------------------
__device__ __forceinline__ unsigned bytemax4(unsigned a, unsigned b) {
  unsigned r = 0;
#pragma unroll
  for (int i = 0; i < 4; ++i) {
    const unsigned aa = (a >> (8 * i)) & 0xFFu;
    const unsigned bb = (b >> (8 * i)) & 0xFFu;
    r |= (aa > bb ? aa : bb) << (8 * i);
  }
  return r;
}

// One thread handles (simplex e, 16-direction chunk c). Each vertex row chunk
// is a single aligned 16-byte load; the vidx table is L2-resident (12.8MB).
template <int NV>
__global__ void __launch_bounds__(256)
wect_simplex_kernel(const int* __restrict__ simp, const float* __restrict__ sw,
                    const unsigned char* __restrict__ vidx,
                    float* __restrict__ ghist, int k, float sign) {
  __shared__ float lhist[NBINS];
  for (int i = threadIdx.x; i < NBINS; i += blockDim.x) lhist[i] = 0.0f;
  __syncthreads();

  const int stride = gridDim.x * blockDim.x;
  const int work = k * 4;  // 4 direction chunks of 16 per simplex
  for (int t = blockIdx.x * blockDim.x + threadIdx.x; t < work; t += stride) {
    const int e = t >> 2;
    const int c = t & 3;
    const float w = sign * sw[e];

    const int v0 = simp[NV * e + 0];
    uint4 mx = *(const uint4*)(vidx + (size_t)v0 * DD + c * 16);
#pragma unroll
    for (int j = 1; j < NV; ++j) {
      const int vj = simp[NV * e + j];
      const uint4 p = *(const uint4*)(vidx + (size_t)vj * DD + c * 16);
      mx.x = bytemax4(mx.x, p.x);
      mx.y = bytemax4(mx.y, p.y);
      mx.z = bytemax4(mx.z, p.z);
      mx.w = bytemax4(mx.w, p.w);
    }

    const int dbase = c * 16;
    const unsigned words[4] = {mx.x, mx.y, mx.z, mx.w};
#pragma unroll
    for (int wi = 0; wi < 4; ++wi) {
#pragma unroll
      for (int b = 0; b < 4; ++b) {
        const int idx = (words[wi] >> (8 * b)) & 0xFF;
        const int dir = dbase + wi * 4 + b;
        atomicAdd(&lhist[dir * HH + idx], w);  // ds_add_f32
      }
    }
  }

  __syncthreads();
  for (int i = threadIdx.x; i < NBINS; i += blockDim.x)
    atomicAdd(&ghist[i], lhist[i]);
}

// ---------------------------------------------------------------------------
__global__ void wect_cumsum_kernel(const float* __restrict__ ghist,
                                   float* __restrict__ out) {
  const int d = threadIdx.x;
  if (d < DD) {
    float run = 0.0f;
    for (int h = 0; h < HH; ++h) {
      run += ghist[d * HH + h];
      out[d * HH + h] = run;
    }
  }
}

// ---------------------------------------------------------------------------
extern "C" void kernel_launch(void* const* d_in, const int* in_sizes, int n_in,
                              void* d_out, int out_size, void* d_ws,
                              size_t ws_size, hipStream_t stream) {
  const float* vc    = (const float*)d_in[0];
  const float* vw    = (const float*)d_in[1];
  const int*   edges = (const int*)d_in[2];
  const float* ew    = (const float*)d_in[3];
  const int*   tris  = (const int*)d_in[4];
  const float* tw    = (const float*)d_in[5];
  const float* dirs  = (const float*)d_in[6];
  const int*   nh    = (const int*)d_in[7];

  const int k0 = in_sizes[0] / 3;
  const int k1 = in_sizes[2] / 2;
  const int k2 = in_sizes[4] / 3;

  char* ws = (char*)d_ws;
  unsigned*      maxslot = (unsigned*)ws;
  float*         ghist   = (float*)(ws + HIST_OFF);
  unsigned char* vidx    = (unsigned char*)(ws + VIDX_OFF);

  wect_init_kernel<<<(NBINS + 255) / 256, 256, 0, stream>>>(ghist, maxslot);
  wect_max_norm_kernel<<<256, 256, 0, stream>>>(vc, k0, maxslot);
  wect_vidx_vertex_kernel<<<512, 256, 0, stream>>>(vc, vw, dirs, maxslot, nh,
                                                   vidx, ghist, k0);
  wect_simplex_kernel<2><<<640, 256, 0, stream>>>(edges, ew, vidx, ghist, k1,
                                                  1.0f);
  wect_simplex_kernel<3><<<640, 256, 0, stream>>>(tris, tw, vidx, ghist, k2,
                                                  -1.0f);
  wect_cumsum_kernel<<<1, 64, 0, stream>>>(ghist, (float*)d_out);
}